// ProgressiveSample_14955076124692
// MI455X (gfx1250) — compile-verified
//
#include <hip/hip_runtime.h>

// Problem constants (from reference):
//   input  [N, C, H, W] fp32,  point/offset [N, P, 2] fp32 (y, x),  GAMMA = 1.0
//   output [N, P, C] fp32
#define N_ 64
#define C_ 256
#define H_ 64
#define W_ 64
#define P_ 1024

// 2 x fp32 pair with only 4-byte alignment guaranteed: single global_load_b64
// on gfx1250 (unaligned-global capable) covering both adjacent x-corners.
typedef float f32x2_a4 __attribute__((ext_vector_type(2), aligned(4)));

// Broadcast a wave-uniform value into an SGPR: point data is per-block
// uniform, so all compares/selects/weights ride the scalar pipe (wave32).
__device__ __forceinline__ float uniform_f32(float v) {
    return __int_as_float(__builtin_amdgcn_readfirstlane(__float_as_int(v)));
}
__device__ __forceinline__ int uniform_i32(int v) {
    return __builtin_amdgcn_readfirstlane(v);
}

__global__ __launch_bounds__(256) void ProgressiveSample_bilinear_kernel(
    const float* __restrict__ inp,     // [N, C, H, W]
    const float* __restrict__ point,   // [N, P, 2] (y, x)
    const float* __restrict__ offs,    // [N, P, 2]
    float* __restrict__ out)           // [N, P, C]
{
    const int c = threadIdx.x;   // channel, 0..255
    const int p = blockIdx.x;    // point,   0..1023 (fastest -> same-image blocks adjacent => L2 locality)
    const int n = blockIdx.y;    // image,   0..63

    // --- wave-uniform point position: pos = point + GAMMA * offset ---
    const size_t pidx = ((size_t)n * P_ + p) * 2;
    const float py = uniform_f32(point[pidx + 0] + offs[pidx + 0]);
    const float px = uniform_f32(point[pidx + 1] + offs[pidx + 1]);

    const float y0f = floorf(py);
    const float x0f = floorf(px);
    const float ly  = py - y0f;
    const float lx  = px - x0f;
    const float hy  = 1.0f - ly;
    const float hx  = 1.0f - lx;
    const int   y0  = uniform_i32((int)y0f);
    const int   x0  = uniform_i32((int)x0f);
    const int   x1  = x0 + 1;
    const int   y1  = y0 + 1;

    // Clamp load coordinates into bounds; zero the WEIGHT of any OOB corner.
    // 0 * finite == 0 reproduces the reference's zero-padding exactly.
    const int sx  = min(max(x0, 0), W_ - 2);   // b64 pair (sx, sx+1) always legal
    const int sy0 = min(max(y0, 0), H_ - 1);
    const int sy1 = min(max(y1, 0), H_ - 1);

    const float wrow0 = (y0 >= 0 && y0 < H_) ? hy : 0.0f;  // SGPR
    const float wrow1 = (y1 >= 0 && y1 < H_) ? ly : 0.0f;  // SGPR
    const float wa    = (x0 >= 0 && x0 < W_) ? hx : 0.0f;  // SGPR
    const float wb    = (x1 >= 0 && x1 < W_) ? lx : 0.0f;  // SGPR

    // Channel plane for this (n, c): 16KB stride between lanes (NCHW).
    const float* __restrict__ plane = inp + ((size_t)(n * C_ + c) * H_) * (size_t)W_;

    // Branch-free gather: both rows' b64 loads issue back-to-back; a single
    // s_wait_loadcnt gates the FMA chain.
    const f32x2_a4 v0 = *(const f32x2_a4*)(plane + (size_t)sy0 * W_ + sx);
    const f32x2_a4 v1 = *(const f32x2_a4*)(plane + (size_t)sy1 * W_ + sx);

    // Recover corner values from the clamped pair (uniform-condition selects):
    //   x0 == sx  -> corner x0 is the low element, else (x0 == W-1) the high.
    //   x1 == sx  -> only when x0 == -1 (then x1 == 0 == sx), else high.
    const bool x0lo = (x0 == sx);
    const bool x1lo = (x1 == sx);
    const float a0 = x0lo ? v0.x : v0.y;
    const float b0 = x1lo ? v0.x : v0.y;
    const float a1 = x0lo ? v1.x : v1.y;
    const float b1 = x1lo ? v1.x : v1.y;

    const float r0 = fmaf(wa, a0, wb * b0);
    const float r1 = fmaf(wa, a1, wb * b1);
    const float acc = fmaf(wrow0, r0, wrow1 * r1);

    // Coalesced store: consecutive lanes -> consecutive channels (128B/wave).
    out[((size_t)n * P_ + p) * C_ + c] = acc;
}

extern "C" void kernel_launch(void* const* d_in, const int* in_sizes, int n_in,
                              void* d_out, int out_size, void* d_ws, size_t ws_size,
                              hipStream_t stream) {
    const float* inp   = (const float*)d_in[0];  // [N, C, H, W]
    const float* point = (const float*)d_in[1];  // [N, P, 2]
    const float* offs  = (const float*)d_in[2];  // [N, P, 2]
    float*       out   = (float*)d_out;          // [N, P, C]

    dim3 grid(P_, N_);          // p fastest => all blocks of image n are temporally adjacent (L2 reuse)
    dim3 block(C_);             // 256 threads = 8 wave32s; one channel per thread
    ProgressiveSample_bilinear_kernel<<<grid, block, 0, stream>>>(inp, point, offs, out);
}